// FeatureGenKerasV2_2095944041143
// MI455X (gfx1250) — compile-verified
//
#include <hip/hip_runtime.h>
#include <hip/hip_bf16.h>
#include <math.h>

typedef __attribute__((ext_vector_type(2))) float v2f;
typedef __attribute__((ext_vector_type(8))) float v8f;

#define FRAME_FLOATS (115 * 3)   // 345 floats per frame
#define NFRAMES_OUT 200

__device__ __forceinline__ float nan0(float v) { return (v == v) ? v : 0.0f; }

// ---------------------------------------------------------------------------
// Phase 0: zero the cross-kernel accumulator in workspace (graph-safe).
// ---------------------------------------------------------------------------
__global__ void zero_kernel(int* p) {
    if (threadIdx.x == 0) p[0] = 0;
}

// ---------------------------------------------------------------------------
// Phase 1: diff = count_nonzero(lefth) - count_nonzero(righth) over ALL
// 100000 frames. Memory-bound: 50.4 MB of reads, ~2.2us at 23.3 TB/s.
// NaN counts as zero (reference applies NaN->0 before the != 0 test).
// ---------------------------------------------------------------------------
__global__ __launch_bounds__(256) void count_kernel(const float* __restrict__ x,
                                                    int* __restrict__ diff,
                                                    int total /* frames*63 */) {
    int acc = 0;
    const int stride = gridDim.x * blockDim.x;
    for (int i = blockIdx.x * blockDim.x + threadIdx.x; i < total; i += stride) {
        unsigned f = (unsigned)i / 63u;
        unsigned r = (unsigned)i - f * 63u;
        const float* base = x + (size_t)f * FRAME_FLOATS;
        float l  = base[120 + r];   // lefth  = x[:, 40:61]  -> floats [120,183)
        float rr = base[282 + r];   // righth = x[:, 94:115] -> floats [282,345)
        acc += ((l == l)  && (l  != 0.0f)) ? 1 : 0;
        acc -= ((rr == rr) && (rr != 0.0f)) ? 1 : 0;
    }
    __shared__ int red[256];
    red[threadIdx.x] = acc;
    __syncthreads();
    for (int s = 128; s > 0; s >>= 1) {
        if (threadIdx.x < s) red[threadIdx.x] += red[threadIdx.x + s];
        __syncthreads();
    }
    if (threadIdx.x == 0) atomicAdd(diff, red[0]);
}

// triu_indices(n, 1) order: (0,1)..(0,n-1),(1,2)...  k -> (i,j)
__device__ __forceinline__ void triu_ij(int k, int n, int& i, int& j) {
    int ii = 0, rem = k, cnt = n - 1;
    while (rem >= cnt) { rem -= cnt; ++ii; --cnt; }
    i = ii;
    j = ii + 1 + rem;
}

// ---------------------------------------------------------------------------
// Phase 2: one block per output frame t (0..199). Builds xfeat[t], xfeat[t+1]
// in LDS, computes the four Gram matrices with v_wmma_f32_16x16x4_f32
// (B operand == A operand VGPRs since G = P * P^T and the A/B lane layouts
// mirror each other), then emits the 1198 feature columns.
// ---------------------------------------------------------------------------
__global__ __launch_bounds__(256) void feat_kernel(const float* __restrict__ x,
                                                   const int* __restrict__ diff,
                                                   float* __restrict__ out) {
    const int t   = blockIdx.x;
    const int tid = threadIdx.x;
    const bool cond = (*diff) > 0;   // lefth has more nonzeros

    __shared__ float xf0[86][3];     // xfeat[t]
    __shared__ float xf1[86][3];     // xfeat[t+1] (for dxyz)
    __shared__ float G[4][32][32];   // Gram matrices of the 4 point groups
    __shared__ float handmask_sh;

    // ---- build xfeat for frames t and t+1 (NaN->0, hand select, x-negate) --
    for (int e = tid; e < 86 * 3 * 2; e += 256) {
        int which = e / 258;              // 0 -> frame t, 1 -> frame t+1
        int pe = e - which * 258;
        int p = pe / 3, c = pe - p * 3;
        int row;
        if (p < 21)      row = cond ? (40 + p) : (94 + p);  // chosen hand
        else if (p < 46) row = 40 + p;                      // pose: 61+(p-21)
        else             row = p - 46;                      // lip
        float v = nan0(x[(size_t)(t + which) * FRAME_FLOATS + row * 3 + c]);
        if (c == 0 && cond) v = -v;       // xc = cond ? -x : x (all 86 points)
        if (which == 0) xf0[p][c] = v; else xf1[p][c] = v;
    }
    __syncthreads();

    // ---- hand_mask: sum of RAW chosen-hand coords != 0 (undo the negation) -
    if (tid == 0) {
        float s = 0.0f;
        for (int p = 0; p < 21; ++p)
            for (int c = 0; c < 3; ++c) {
                float v = xf0[p][c];
                if (c == 0 && cond) v = -v;   // recover raw value
                s += v;
            }
        handmask_sh = (s != 0.0f) ? 1.0f : 0.0f;
    }

    // ---- Gram matrices via WMMA: one group per wave (waves 0..3) ----------
    const int wave = tid >> 5;
    const int lane = tid & 31;
    if (wave < 4) {
        int base, npts, nco;
        if (wave == 0)      { base = 0;  npts = 21; nco = 3; }  // hand, 3D
        else if (wave == 1) { base = 21; npts = 25; nco = 2; }  // pose, 2D
        else if (wave == 2) { base = 46; npts = 20; nco = 2; }  // outer lip
        else                { base = 66; npts = 20; nco = 2; }  // inner lip

        // A operand 16x4 f32: lane<16 -> (M=lane, K={0,1}); lane>=16 -> (M=lane-16, K={2,3})
        const int m  = lane & 15;
        const int k0 = (lane < 16) ? 0 : 2;
        v2f A0, A1;
        {
            int p = base + m;                      // tile0 rows always valid (npts>=20>16... m<16)
            A0.x = (k0     < nco) ? xf0[p][k0]     : 0.0f;
            A0.y = (k0 + 1 < nco) ? xf0[p][k0 + 1] : 0.0f;
        }
        {
            int  p  = base + 16 + m;
            bool ok = (16 + m) < npts;
            A1.x = (ok && k0     < nco) ? xf0[p][k0]     : 0.0f;
            A1.y = (ok && k0 + 1 < nco) ? xf0[p][k0 + 1] : 0.0f;
        }
        v8f z = {};
        // B layout mirrors A layout (lane->N, lane-half->K pair), so for
        // G = P*P^T the B operand registers are identical to A's.
        v8f D00 = __builtin_amdgcn_wmma_f32_16x16x4_f32(false, A0, false, A0, (short)0, z, false, false);
        v8f D01 = __builtin_amdgcn_wmma_f32_16x16x4_f32(false, A0, false, A1, (short)0, z, false, false);
        v8f D11 = __builtin_amdgcn_wmma_f32_16x16x4_f32(false, A1, false, A1, (short)0, z, false, false);

        // D layout: vgpr v, lane l -> M = v + 8*(l>=16), N = l%16
        const int M0 = (lane < 16) ? 0 : 8;
        const int N  = lane & 15;
#pragma unroll
        for (int v = 0; v < 8; ++v) {
            G[wave][M0 + v][N]            = D00[v];
            G[wave][M0 + v][16 + N]       = D01[v];   // upper-right block (i<16<=j)
            G[wave][16 + M0 + v][16 + N]  = D11[v];
        }
        // lower-left block never read: triu needs only i<j and the diagonal.
    }
    __syncthreads();

    // ---- emit the 1198 feature columns ------------------------------------
    const float hm = handmask_sh;
    float* o = out + (size_t)t * 1198;
    for (int col = tid; col < 1198; col += 256) {
        float val;
        if (col < 63) {                               // xfeat[:, :21, :3]
            int p = col / 3, c = col - p * 3;
            val = xf0[p][c];
        } else if (col < 113) {                       // xfeat[:, 21:46, :2]
            int j = col - 63; int p = 21 + (j >> 1);
            val = xf0[p][j & 1];
        } else if (col < 153) {                       // xfeat[:, 46:66, :2]
            int j = col - 113; int p = 46 + (j >> 1);
            val = xf0[p][j & 1];
        } else if (col < 216) {                       // dxyz[:, :21, :3]
            int j = col - 153; int p = j / 3, c = j - p * 3;
            val = xf0[p][c] - xf1[p][c];
        } else if (col < 266) {                       // dxyz[:, 21:46, :2]
            int j = col - 216; int p = 21 + (j >> 1); int c = j & 1;
            val = xf0[p][c] - xf1[p][c];
        } else if (col < 306) {                       // dxyz[:, 46:66, :2]
            int j = col - 266; int p = 46 + (j >> 1); int c = j & 1;
            val = xf0[p][c] - xf1[p][c];
        } else if (col < 516) {                       // hdist (210)
            int i, j; triu_ij(col - 306, 21, i, j);
            float d2 = G[0][i][i] + G[0][j][j] - 2.0f * G[0][i][j];
            val = sqrtf(fmaxf(d2, 0.0f));
        } else if (col < 816) {                       // pdist (300)
            int i, j; triu_ij(col - 516, 25, i, j);
            float d2 = G[1][i][i] + G[1][j][j] - 2.0f * G[1][i][j];
            val = sqrtf(fmaxf(d2, 0.0f));
        } else if (col < 1006) {                      // oldist (190)
            int i, j; triu_ij(col - 816, 20, i, j);
            float d2 = G[2][i][i] + G[2][j][j] - 2.0f * G[2][i][j];
            val = sqrtf(fmaxf(d2, 0.0f));
        } else if (col < 1196) {                      // ildist (190)
            int i, j; triu_ij(col - 1006, 20, i, j);
            float d2 = G[3][i][i] + G[3][j][j] - 2.0f * G[3][i][j];
            val = sqrtf(fmaxf(d2, 0.0f));
        } else if (col == 1196) {                     // hand_mask
            val = hm;
        } else {                                      // token_type_ids
            val = hm + 1.0f;
        }
        o[col] = val;
    }
}

// ---------------------------------------------------------------------------
extern "C" void kernel_launch(void* const* d_in, const int* in_sizes, int n_in,
                              void* d_out, int out_size, void* d_ws, size_t ws_size,
                              hipStream_t stream) {
    const float* x = (const float*)d_in[0];
    float* out = (float*)d_out;
    int* diff = (int*)d_ws;

    const int T = in_sizes[0] / FRAME_FLOATS;   // 100000 frames
    const int total = T * 63;                   // elements per hand slice

    zero_kernel<<<1, 32, 0, stream>>>(diff);
    count_kernel<<<2048, 256, 0, stream>>>(x, diff, total);
    feat_kernel<<<NFRAMES_OUT, 256, 0, stream>>>(x, diff, out);
}